// Global_Local_72456098283774
// MI455X (gfx1250) — compile-verified
//
#include <hip/hip_runtime.h>
#include <math.h>

typedef float v2f __attribute__((ext_vector_type(2)));
typedef float v8f __attribute__((ext_vector_type(8)));

#define LSEQ 5880
#define NHEAD 16
#define NB 2
#define DM 12
#define DI 24
#define DS 16
#define XDIM 33
#define NTILE 368   // ceil(5880/16)
#define NCH 60      // scan chunks
#define CHL 98      // 60*98 == 5880 exactly

#if __has_builtin(__builtin_amdgcn_global_load_async_to_lds_b32)
#define USE_ASYNC_LDS 1
#endif

// D = A(16x4,f32) * B(4x16,f32) + C(16x16,f32), full wave32 op.
__device__ __forceinline__ v8f wmma4(v2f a, v2f b, v8f c) {
  return __builtin_amdgcn_wmma_f32_16x16x4_f32(false, a, false, b, (short)0, c, false, false);
}

__device__ __forceinline__ float siluf(float x) { return x / (1.f + expf(-x)); }

// Contiguous global -> LDS copy. Uses CDNA5 async-to-LDS path when available
// (ASYNCcnt-tracked, no VGPR round trip), else plain copy.
__device__ __forceinline__ void ldscpy(float* l, const float* g, int n, int tid, int stride) {
  for (int i = tid; i < n; i += stride) {
#ifdef USE_ASYNC_LDS
    __builtin_amdgcn_global_load_async_to_lds_b32(
        (__attribute__((address_space(1))) int*)(float*)(g + i),
        (__attribute__((address_space(3))) int*)(l + i), 0, 0);
#else
    l[i] = g[i];
#endif
  }
}

__device__ __forceinline__ void ldscpy_join() {
#ifdef USE_ASYNC_LDS
#if __has_builtin(__builtin_amdgcn_s_wait_asynccnt)
  __builtin_amdgcn_s_wait_asynccnt(0);
#else
  asm volatile("s_wait_asynccnt 0x0" ::: "memory");
#endif
#endif
  __syncthreads();
}

// ---------------------------------------------------------------- K1: gather xe
__global__ void k_gather(const float* __restrict__ x0, const float* __restrict__ x1,
                         const float* __restrict__ x2, const float* __restrict__ x3,
                         float* __restrict__ xe) {
  size_t i = (size_t)blockIdx.x * 256 + threadIdx.x;
  size_t tot = (size_t)NB * LSEQ * 96;
  if (i >= tot) return;
  int c = (int)(i % 96);
  size_t rest = i / 96;
  int l = (int)(rest % LSEQ);
  int b = (int)(rest / LSEQ);
  float v;
  if (l < 3136)       v = x0[((size_t)b * 3136 + l) * 96 + c];
  else if (l < 4704)  v = x1[((size_t)b * 1568 + (l - 3136)) * 96 + c];
  else if (l < 5488)  v = x2[((size_t)b * 784  + (l - 4704)) * 96 + c];
  else                v = x3[((size_t)b * 392  + (l - 5488)) * 96 + c];
  xe[i] = v;
}

// ---------------------------------------------------------------- K2: in_proj (WMMA) + silu(z)
__global__ void k_inproj(const float* __restrict__ xe, const float* __restrict__ Win,
                         float* __restrict__ xcr, float* __restrict__ zs) {
  int tile = blockIdx.x, b = blockIdx.y, h = blockIdx.z;
  int lane = threadIdx.x;
  int half = lane >> 4, ln = lane & 15;
  int l = tile * 16 + ln;
  bool valid = (l < LSEQ);
  int lsrc = (h < 8) ? l : (LSEQ - 1 - l);
  if (!valid) lsrc = 0;
  int cbase = ((h < 8) ? h : (h - 8)) * DM;
  const float* xrow = xe + ((size_t)b * LSEQ + lsrc) * 96 + cbase;
  const float* Wh = Win + (size_t)h * 48 * DM;   // [n][k]
  __builtin_prefetch(Wh, 0, 3);
  v2f a[3];
#pragma unroll
  for (int kb = 0; kb < 3; kb++) {
    int k0 = kb * 4 + 2 * half;
    a[kb].x = valid ? xrow[k0]     : 0.f;
    a[kb].y = valid ? xrow[k0 + 1] : 0.f;
  }
  size_t hb = (size_t)h * NB + b;
  int row0 = tile * 16 + 8 * half;
  bool fullt = (row0 + 8 <= LSEQ);
#pragma unroll
  for (int nt = 0; nt < 3; nt++) {
    v8f c = {0.f,0.f,0.f,0.f,0.f,0.f,0.f,0.f};
#pragma unroll
    for (int kb = 0; kb < 3; kb++) {
      int k0 = kb * 4 + 2 * half;
      int n = nt * 16 + ln;
      v2f bf; bf.x = Wh[n * DM + k0]; bf.y = Wh[n * DM + k0 + 1];
      c = wmma4(a[kb], bf, c);
    }
    int col = nt * 16 + ln;
    bool isx = (col < DI);
    float* p = isx ? (xcr + (hb * LSEQ + row0) * DI + col)
                   : (zs  + (hb * LSEQ + row0) * DI + (col - DI));
    if (fullt) {
#pragma unroll
      for (int i = 0; i < 8; i++) { float v = c[i]; p[(size_t)i * DI] = isx ? v : siluf(v); }
    } else {
      for (int i = 0; i < 8; i++)
        if (row0 + i < LSEQ) { float v = c[i]; p[(size_t)i * DI] = isx ? v : siluf(v); }
    }
  }
}

// ---------------------------------------------------------------- K3: causal conv4+silu, x_proj (WMMA), softplus delta
__global__ void k_convxproj(const float* __restrict__ xcr, const float* __restrict__ convw,
                            const float* __restrict__ convb, const float* __restrict__ Wx,
                            const float* __restrict__ Wdt, const float* __restrict__ bdt,
                            float* __restrict__ xcs, float* __restrict__ delta,
                            float* __restrict__ Bm, float* __restrict__ Cm) {
  __shared__ float sxc[16][DI];
  __shared__ float sdt[16];
  int tile = blockIdx.x, b = blockIdx.y, h = blockIdx.z;
  int lane = threadIdx.x;
  int half = lane >> 4, ln = lane & 15;
  size_t hb = (size_t)h * NB + b;
  for (int idx = lane; idx < 16 * DI; idx += 32) {
    int r = idx / DI, d = idx % DI;
    int l = tile * 16 + r;
    float acc = 0.f;
    if (l < LSEQ) {
      acc = convb[h * DI + d];
#pragma unroll
      for (int k = 0; k < 4; k++) {
        int lp = l - 3 + k;
        if (lp >= 0) acc += xcr[(hb * LSEQ + lp) * DI + d] * convw[((size_t)h * DI + d) * 4 + k];
      }
      acc = siluf(acc);
      xcs[(hb * LSEQ + l) * DI + d] = acc;
    }
    sxc[r][d] = acc;
  }
  __syncthreads();
  v2f a[6];
#pragma unroll
  for (int kb = 0; kb < 6; kb++) {
    int k0 = kb * 4 + 2 * half;
    a[kb].x = sxc[ln][k0]; a[kb].y = sxc[ln][k0 + 1];
  }
  const float* Wxh = Wx + (size_t)h * XDIM * DI;  // [n][k]
  __builtin_prefetch(Wxh, 0, 3);
  int row0 = tile * 16 + 8 * half;
  bool fullt = (row0 + 8 <= LSEQ);
#pragma unroll
  for (int nt = 0; nt < 3; nt++) {
    v8f c = {0.f,0.f,0.f,0.f,0.f,0.f,0.f,0.f};
#pragma unroll
    for (int kb = 0; kb < 6; kb++) {
      int k0 = kb * 4 + 2 * half;
      int n = nt * 16 + ln;
      v2f bf;
      if (n < XDIM) { bf.x = Wxh[n * DI + k0]; bf.y = Wxh[n * DI + k0 + 1]; }
      else          { bf.x = 0.f; bf.y = 0.f; }
      c = wmma4(a[kb], bf, c);
    }
    int col = nt * 16 + ln;
    float* p = nullptr;
    if (col >= 1 && col < 1 + DS)            p = Bm + (hb * LSEQ + row0) * DS + (col - 1);
    else if (col >= 1 + DS && col < 1+2*DS)  p = Cm + (hb * LSEQ + row0) * DS + (col - 1 - DS);
    if (p) {
      if (fullt) {
#pragma unroll
        for (int i = 0; i < 8; i++) p[(size_t)i * DS] = c[i];
      } else {
        for (int i = 0; i < 8; i++) if (row0 + i < LSEQ) p[(size_t)i * DS] = c[i];
      }
    }
    if (col == 0) {
#pragma unroll
      for (int i = 0; i < 8; i++) {
        int r = i + 8 * half;
        sdt[r] = (tile * 16 + r < LSEQ) ? c[i] : 0.f;
      }
    }
  }
  __syncthreads();
  for (int idx = lane; idx < 16 * DI; idx += 32) {
    int r = idx / DI, d = idx % DI;
    int l = tile * 16 + r;
    if (l >= LSEQ) continue;
    float x = sdt[r] * Wdt[h * DI + d] + bdt[h * DI + d];
    delta[(hb * LSEQ + l) * DI + d] = (x > 20.f) ? x : log1pf(expf(x));
  }
}

// ---------------------------------------------------------------- K4a: per-chunk local scan (P = exp(A*Σδ), H = local end state)
__global__ void k_scanA(const float* __restrict__ delta, const float* __restrict__ xcs,
                        const float* __restrict__ Bm, const float* __restrict__ Alog,
                        float* __restrict__ P, float* __restrict__ H) {
  __shared__ float sdel[CHL * DI];
  __shared__ float sx[CHL * DI];
  __shared__ float sB[CHL * DS];
  int ch = blockIdx.x, b = blockIdx.y, h = blockIdx.z;
  size_t hb = (size_t)h * NB + b;
  int tid = threadIdx.x;
  int l0 = ch * CHL;
  ldscpy(sdel, delta + (hb * LSEQ + l0) * DI, CHL * DI, tid, 384);
  ldscpy(sx,   xcs   + (hb * LSEQ + l0) * DI, CHL * DI, tid, 384);
  ldscpy(sB,   Bm    + (hb * LSEQ + l0) * DS, CHL * DS, tid, 384);
  ldscpy_join();
  int d = tid >> 4, s = tid & 15;
  float A = -expf(Alog[((size_t)h * DI + d) * DS + s]);
  float hst = 0.f, sd = 0.f;
  for (int r = 0; r < CHL; r++) {
    float de = sdel[r * DI + d];
    float aa = expf(de * A);
    hst = aa * hst + de * sB[r * DS + s] * sx[r * DI + d];
    sd += de;
  }
  size_t o = (hb * NCH + ch) * (DI * DS) + tid;
  P[o] = expf(A * sd);
  H[o] = hst;
}

// ---------------------------------------------------------------- K4b: combine 60 chunk aggregates -> per-chunk initial state
__global__ void k_scanB(const float* __restrict__ P, const float* __restrict__ H,
                        float* __restrict__ Hin) {
  size_t hb = blockIdx.x;
  int tid = threadIdx.x;
  float hst = 0.f;
  for (int c = 0; c < NCH; c++) {
    size_t o = (hb * NCH + c) * (DI * DS) + tid;
    Hin[o] = hst;
    hst = P[o] * hst + H[o];
  }
}

// ---------------------------------------------------------------- K4c: final scan + C-einsum + D*x + silu(z) gate
__global__ void k_scanC(const float* __restrict__ delta, const float* __restrict__ xcs,
                        const float* __restrict__ zs, const float* __restrict__ Bm,
                        const float* __restrict__ Cm, const float* __restrict__ Alog,
                        const float* __restrict__ Dp, const float* __restrict__ Hin,
                        float* __restrict__ ya) {
  __shared__ float sdel[CHL * DI];
  __shared__ float sx[CHL * DI];
  __shared__ float sz[CHL * DI];
  __shared__ float sB[CHL * DS];
  __shared__ float sC[CHL * DS];
  int ch = blockIdx.x, b = blockIdx.y, h = blockIdx.z;
  size_t hb = (size_t)h * NB + b;
  int tid = threadIdx.x;
  int l0 = ch * CHL;
  ldscpy(sdel, delta + (hb * LSEQ + l0) * DI, CHL * DI, tid, 384);
  ldscpy(sx,   xcs   + (hb * LSEQ + l0) * DI, CHL * DI, tid, 384);
  ldscpy(sz,   zs    + (hb * LSEQ + l0) * DI, CHL * DI, tid, 384);
  ldscpy(sB,   Bm    + (hb * LSEQ + l0) * DS, CHL * DS, tid, 384);
  ldscpy(sC,   Cm    + (hb * LSEQ + l0) * DS, CHL * DS, tid, 384);
  ldscpy_join();
  int d = tid >> 4, s = tid & 15;
  float A = -expf(Alog[((size_t)h * DI + d) * DS + s]);
  float Dd = Dp[h * DI + d];
  float hst = Hin[(hb * NCH + ch) * (DI * DS) + tid];
  for (int r = 0; r < CHL; r++) {
    float de = sdel[r * DI + d];
    float aa = expf(de * A);
    hst = aa * hst + de * sB[r * DS + s] * sx[r * DI + d];
    float ct = hst * sC[r * DS + s];
    ct += __shfl_xor(ct, 1, 16);
    ct += __shfl_xor(ct, 2, 16);
    ct += __shfl_xor(ct, 4, 16);
    ct += __shfl_xor(ct, 8, 16);
    if (s == 0) {
      float y = ct + Dd * sx[r * DI + d];
      ya[(hb * LSEQ + l0 + r) * DI + d] = y * sz[r * DI + d];
    }
  }
}

// ---------------------------------------------------------------- K5: out_proj (WMMA)
__global__ void k_outproj(const float* __restrict__ ya, const float* __restrict__ Wout,
                          float* __restrict__ yo) {
  int tile = blockIdx.x, b = blockIdx.y, h = blockIdx.z;
  int lane = threadIdx.x;
  int half = lane >> 4, ln = lane & 15;
  size_t hb = (size_t)h * NB + b;
  int l = tile * 16 + ln;
  bool valid = (l < LSEQ);
  const float* yrow = ya + (hb * LSEQ + (valid ? l : 0)) * DI;
  v2f a[6];
#pragma unroll
  for (int kb = 0; kb < 6; kb++) {
    int k0 = kb * 4 + 2 * half;
    a[kb].x = valid ? yrow[k0]     : 0.f;
    a[kb].y = valid ? yrow[k0 + 1] : 0.f;
  }
  const float* Wh = Wout + (size_t)h * DM * DI;  // [n][k]
  v8f c = {0.f,0.f,0.f,0.f,0.f,0.f,0.f,0.f};
#pragma unroll
  for (int kb = 0; kb < 6; kb++) {
    int k0 = kb * 4 + 2 * half;
    v2f bf;
    if (ln < DM) { bf.x = Wh[ln * DI + k0]; bf.y = Wh[ln * DI + k0 + 1]; }
    else         { bf.x = 0.f; bf.y = 0.f; }
    c = wmma4(a[kb], bf, c);
  }
  int row0 = tile * 16 + 8 * half;
  if (ln < DM) {
    float* p = yo + (hb * LSEQ + row0) * DM + ln;
    if (row0 + 8 <= LSEQ) {
#pragma unroll
      for (int i = 0; i < 8; i++) p[(size_t)i * DM] = c[i];
    } else {
      for (int i = 0; i < 8; i++) if (row0 + i < LSEQ) p[(size_t)i * DM] = c[i];
    }
  }
}

// ---------------------------------------------------------------- K6: combine heads + layernorm
__global__ void k_ln(const float* __restrict__ xe, const float* __restrict__ yo,
                     const float* __restrict__ g, const float* __restrict__ be,
                     float* __restrict__ xn) {
  __shared__ float sv[96], sq[96];
  __shared__ float smu, srv;
  int row = blockIdx.x;
  int b = row / LSEQ, l = row % LSEQ;
  int c = threadIdx.x;
  int h1 = c / DM, j1 = c % DM;
  int cp = 95 - c;
  int h2 = 8 + cp / DM, j2 = cp % DM;
  float v = 2.f * xe[((size_t)b * LSEQ + l) * 96 + c]
          + yo[(((size_t)h1 * NB + b) * LSEQ + l) * DM + j1]
          + yo[(((size_t)h2 * NB + b) * LSEQ + l) * DM + j2];
  sv[c] = v; sq[c] = v * v;
  __syncthreads();
  if (c < 32) {
    float s1 = sv[c] + sv[c + 32] + sv[c + 64];
    float q1 = sq[c] + sq[c + 32] + sq[c + 64];
    for (int m = 1; m < 32; m <<= 1) { s1 += __shfl_xor(s1, m, 32); q1 += __shfl_xor(q1, m, 32); }
    if (c == 0) { float mu = s1 / 96.f; smu = mu; srv = rsqrtf(q1 / 96.f - mu * mu + 1e-5f); }
  }
  __syncthreads();
  xn[((size_t)b * LSEQ + l) * 96 + c] = (v - smu) * srv * g[c] + be[c];
}

// ---------------------------------------------------------------- K7a: ConvGLU fc1 (WMMA), split xh / v
__global__ void k_glu_f1(const float* __restrict__ xn, const float* __restrict__ f1w,
                         const float* __restrict__ f1b, float* __restrict__ xh,
                         float* __restrict__ vv, int segstart, int Nt, int dim, int hid) {
  int tile = blockIdx.x;
  int lane = threadIdx.x;
  int half = lane >> 4, ln = lane & 15;
  int toks = NB * Nt;
  int t = tile * 16 + ln;
  bool valid = (t < toks);
  int ts = valid ? t : 0;
  int b = ts / Nt, n = ts % Nt;
  const float* xrow = xn + ((size_t)b * LSEQ + segstart) * 96 + (size_t)n * dim;
  __builtin_prefetch(xrow, 0, 3);
  __builtin_prefetch(f1w, 0, 3);
  int hid2 = hid / 2;
  int ksteps = dim >> 2;
  int ntiles = (hid + 15) >> 4;
  int tt0 = tile * 16 + 8 * half;
  for (int nt = 0; nt < ntiles; nt++) {
    v8f c = {0.f,0.f,0.f,0.f,0.f,0.f,0.f,0.f};
    for (int kb = 0; kb < ksteps; kb++) {
      int k0 = kb * 4 + 2 * half;
      v2f a;
      a.x = valid ? xrow[k0]     : 0.f;
      a.y = valid ? xrow[k0 + 1] : 0.f;
      int nn = nt * 16 + ln;
      v2f bf;
      if (nn < hid) { bf.x = f1w[(size_t)nn * dim + k0]; bf.y = f1w[(size_t)nn * dim + k0 + 1]; }
      else          { bf.x = 0.f; bf.y = 0.f; }
      c = wmma4(a, bf, c);
    }
    int col = nt * 16 + ln;
    if (col < hid) {
      float bia = f1b[col];
      float* p = (col < hid2) ? (xh + (size_t)tt0 * hid2 + col)
                              : (vv + (size_t)tt0 * hid2 + (col - hid2));
      if (tt0 + 8 <= toks) {
#pragma unroll
        for (int i = 0; i < 8; i++) p[(size_t)i * hid2] = c[i] + bia;
      } else {
        for (int i = 0; i < 8; i++) if (tt0 + i < toks) p[(size_t)i * hid2] = c[i] + bia;
      }
    }
  }
}

// ---------------------------------------------------------------- K7b: depthwise3x3 + gelu*v + fc2 + residuals -> output
__global__ void k_glu_f2(const float* __restrict__ xe, const float* __restrict__ xn,
                         const float* __restrict__ xh, const float* __restrict__ vv,
                         const float* __restrict__ dww, const float* __restrict__ dwb,
                         const float* __restrict__ f2w, const float* __restrict__ f2b,
                         float* __restrict__ out, int segstart, int outoff,
                         int Hh, int Ww, int dim, int hid) {
  __shared__ float st[24];
  int t = blockIdx.x;
  int Nt = Hh * Ww;
  int b = t / Nt, n = t % Nt;
  int y = n / Ww, x = n % Ww;
  int hid2 = hid / 2;
  int tid = threadIdx.x;
  if (tid < hid2) {
    float acc = dwb[tid];
#pragma unroll
    for (int dy = -1; dy <= 1; dy++)
#pragma unroll
      for (int dx = -1; dx <= 1; dx++) {
        int yy = y + dy, xx = x + dx;
        if (yy >= 0 && yy < Hh && xx >= 0 && xx < Ww)
          acc += xh[((size_t)b * Nt + yy * Ww + xx) * hid2 + tid] * dww[tid * 9 + (dy + 1) * 3 + (dx + 1)];
      }
    float ge = 0.5f * acc * (1.f + erff(acc * 0.70710678118654752f));
    st[tid] = ge * vv[(size_t)t * hid2 + tid];
  }
  __syncthreads();
  int seglen = Nt * dim / 96;
  for (int c = tid; c < dim; c += blockDim.x) {
    float acc = f2b[c];
    for (int ch = 0; ch < hid2; ch++) acc += st[ch] * f2w[(size_t)c * hid2 + ch];
    int flat = n * dim + c;
    int l96 = flat / 96, c96 = flat % 96;
    size_t gi = ((size_t)b * LSEQ + segstart + l96) * 96 + c96;
    // y = xe + xn + (fc2 + res(xn)) = xe + 2*xn + fc2
    out[(size_t)outoff + ((size_t)b * seglen + l96) * 96 + c96] = xe[gi] + 2.f * xn[gi] + acc;
  }
}

// ---------------------------------------------------------------- launcher
extern "C" void kernel_launch(void* const* d_in, const int* in_sizes, int n_in,
                              void* d_out, int out_size, void* d_ws, size_t ws_size,
                              hipStream_t stream) {
  (void)in_sizes; (void)n_in; (void)out_size; (void)ws_size;
  const float* x0    = (const float*)d_in[0];
  const float* x1    = (const float*)d_in[1];
  const float* x2    = (const float*)d_in[2];
  const float* x3    = (const float*)d_in[3];
  const float* Win   = (const float*)d_in[4];
  const float* convw = (const float*)d_in[5];
  const float* convb = (const float*)d_in[6];
  const float* Wx    = (const float*)d_in[7];
  const float* Wdt   = (const float*)d_in[8];
  const float* bdt   = (const float*)d_in[9];
  const float* Alog  = (const float*)d_in[10];
  const float* Dp    = (const float*)d_in[11];
  const float* Wout  = (const float*)d_in[12];
  const float* lng   = (const float*)d_in[13];
  const float* lnb   = (const float*)d_in[14];
  float* out = (float*)d_out;
  float* ws  = (float*)d_ws;

  const size_t SZ_XE = (size_t)NB * LSEQ * 96;
  const size_t SZ24  = (size_t)NHEAD * NB * LSEQ * DI;
  const size_t SZ16  = (size_t)NHEAD * NB * LSEQ * DS;
  const size_t SZPH  = (size_t)NHEAD * NB * NCH * DI * DS;

  size_t o = 0;
  float* xe    = ws + o; o += SZ_XE;
  float* xcr   = ws + o; o += SZ24;   // reused as ya after K3
  float* zsb   = ws + o; o += SZ24;
  float* xcs   = ws + o; o += SZ24;
  float* delta = ws + o; o += SZ24;   // reused as yo after K4c
  float* Bmw   = ws + o; o += SZ16;
  float* Cmw   = ws + o; o += SZ16;
  float* xnw   = ws + o; o += SZ_XE;
  float* Pw    = ws + o; o += SZPH;
  float* Hw    = ws + o; o += SZPH;
  float* Hinw  = ws + o; o += SZPH;
  float* xhw   = ws + o; o += (size_t)18816;
  float* vvw   = ws + o; o += (size_t)18816;
  float* ya = xcr;
  float* yo = delta;

  k_gather<<<dim3((unsigned)((SZ_XE + 255) / 256)), 256, 0, stream>>>(x0, x1, x2, x3, xe);

  dim3 gT(NTILE, NB, NHEAD);
  k_inproj<<<gT, 32, 0, stream>>>(xe, Win, xcr, zsb);
  k_convxproj<<<gT, 32, 0, stream>>>(xcr, convw, convb, Wx, Wdt, bdt, xcs, delta, Bmw, Cmw);

  dim3 gC(NCH, NB, NHEAD);
  k_scanA<<<gC, 384, 0, stream>>>(delta, xcs, Bmw, Alog, Pw, Hw);
  k_scanB<<<NHEAD * NB, 384, 0, stream>>>(Pw, Hw, Hinw);
  k_scanC<<<gC, 384, 0, stream>>>(delta, xcs, zsb, Bmw, Cmw, Alog, Dp, Hinw, ya);

  k_outproj<<<gT, 32, 0, stream>>>(ya, Wout, yo);
  k_ln<<<NB * LSEQ, 96, 0, stream>>>(xe, yo, lng, lnb, xnw);

  const int segs[4] = {0, 3136, 4704, 5488};
  const int hw[4]   = {56, 28, 14, 7};
  const int dims[4] = {96, 192, 384, 768};
  const int hids[4] = {6, 12, 24, 48};
  const int ooff[4] = {0, 602112, 903168, 1053696};
  for (int i = 0; i < 4; i++) {
    const float* f1w = (const float*)d_in[15 + i * 6 + 0];
    const float* f1b = (const float*)d_in[15 + i * 6 + 1];
    const float* dww = (const float*)d_in[15 + i * 6 + 2];
    const float* dwb = (const float*)d_in[15 + i * 6 + 3];
    const float* f2w = (const float*)d_in[15 + i * 6 + 4];
    const float* f2b = (const float*)d_in[15 + i * 6 + 5];
    int Nt = hw[i] * hw[i];
    int toks = NB * Nt;
    int tt = (toks + 15) / 16;
    k_glu_f1<<<tt, 32, 0, stream>>>(xnw, f1w, f1b, xhw, vvw, segs[i], Nt, dims[i], hids[i]);
    k_glu_f2<<<toks, 128, 0, stream>>>(xe, xnw, xhw, vvw, dww, dwb, f2w, f2b, out,
                                       segs[i], ooff[i], hw[i], hw[i], dims[i], hids[i]);
  }
}